// SimplicialGeometryBias_85667417686316
// MI455X (gfx1250) — compile-verified
//
#include <hip/hip_runtime.h>
#include <math.h>

typedef __attribute__((ext_vector_type(2))) float v2f;
typedef __attribute__((ext_vector_type(8))) float v8f;

#define NATOMS 320
#define MTILES 20              // 320 / 16
#define WS_BATCH_STRIDE 128    // floats per batch in workspace

// ---------------------------------------------------------------------------
// Setup: per batch, compute Gram matrix G, cell_scale, and per-offset rows
//   w_k = 2*G*off_k (3 floats), c_k = off_k^T G off_k  (27 offsets)
// Workspace layout per batch (floats):
//   [0..8]  = G row-major, [9] = cell_scale, [16 + 4k .. 16+4k+3] = wx,wy,wz,c
// ---------------------------------------------------------------------------
__global__ void geom_setup_kernel(const float* __restrict__ lattice,
                                  float* __restrict__ ws, int B)
{
    int b = blockIdx.x;
    int lane = threadIdx.x;   // block of 32
    if (b >= B) return;

    const float* L = lattice + b * 6;
    float ll0 = fminf(fmaxf(L[0], -5.f), 5.f);
    float ll1 = fminf(fmaxf(L[1], -5.f), 5.f);
    float ll2 = fminf(fmaxf(L[2], -5.f), 5.f);
    float ca  = fminf(fmaxf(L[3], -0.9999f), 0.9999f);
    float cb  = fminf(fmaxf(L[4], -0.9999f), 0.9999f);
    float cg  = fminf(fmaxf(L[5], -0.9999f), 0.9999f);
    float a = expf(ll0), bl = expf(ll1), c = expf(ll2);

    float g00 = a * a,       g01 = a * bl * cg, g02 = a * c * cb;
    float g11 = bl * bl,     g12 = bl * c * ca;
    float g22 = c * c;

    float* wb = ws + (size_t)b * WS_BATCH_STRIDE;

    if (lane == 0) {
        wb[0] = g00; wb[1] = g01; wb[2] = g02;
        wb[3] = g01; wb[4] = g11; wb[5] = g12;
        wb[6] = g02; wb[7] = g12; wb[8] = g22;
        wb[9] = fmaxf((a + bl + c) * (1.0f / 3.0f), 1e-6f);
    }
    if (lane < 27) {
        float ox = (float)(lane / 9) - 1.0f;
        float oy = (float)((lane / 3) % 3) - 1.0f;
        float oz = (float)(lane % 3) - 1.0f;
        float gx = g00 * ox + g01 * oy + g02 * oz;
        float gy = g01 * ox + g11 * oy + g12 * oz;
        float gz = g02 * ox + g12 * oy + g22 * oz;
        float* row = wb + 16 + 4 * lane;
        row[0] = 2.0f * gx;
        row[1] = 2.0f * gy;
        row[2] = 2.0f * gz;
        row[3] = ox * gx + oy * gy + oz * gz;
    }
}

// ---------------------------------------------------------------------------
// Main: one wave32 per (b, n, m-tile of 16). Two V_WMMA_F32_16X16X4_F32 ops
// evaluate dist2 for 27 PBC images x 16 pairs; min/argmin reduced in-lane
// over the 8 D VGPRs, then across wave halves with a lane-16 xor shuffle.
// ---------------------------------------------------------------------------
__global__ void geom_main_kernel(const float* __restrict__ coords,
                                 const float* __restrict__ ws,
                                 float* __restrict__ out, int B)
{
    const int wavesPerBlock = blockDim.x >> 5;
    int w    = blockIdx.x * wavesPerBlock + (threadIdx.x >> 5);
    int lane = threadIdx.x & 31;

    const int tilesPerBatch = NATOMS * MTILES;
    int b   = w / tilesPerBatch;
    if (b >= B) return;                 // grid sized exactly; no divergence
    int rem = w - b * tilesPerBatch;
    int n   = rem / MTILES;
    int mt  = rem - n * MTILES;
    int m0  = mt * 16;

    int col  = lane & 15;
    int half = lane >> 4;

    const float* wb = ws + (size_t)b * WS_BATCH_STRIDE;
    float g00 = wb[0], g01 = wb[1], g02 = wb[2];
    float g11 = wb[4], g12 = wb[5], g22 = wb[8];
    float cell_scale = wb[9];

    const float* cn = coords + ((size_t)b * NATOMS + n) * 3;
    const float* cm = coords + ((size_t)b * NATOMS + m0 + col) * 3;
    float dx = cn[0] - cm[0];
    float dy = cn[1] - cm[1];
    float dz = cn[2] - cm[2];

    // q = delta^T G delta (per column)
    float gdx = g00 * dx + g01 * dy + g02 * dz;
    float gdy = g01 * dx + g11 * dy + g12 * dz;
    float gdz = g02 * dx + g12 * dy + g22 * dz;
    float q = dx * gdx + dy * gdy + dz * gdz;

    // B fragment (4x16): VGPR0 = K0(dx)/K2(dz), VGPR1 = K1(dy)/K3(1)
    v2f Bf;
    Bf.x = half ? dz : dx;
    Bf.y = half ? 1.0f : dy;

    // A fragments (16x4): lane%16 = row M = offset index; VGPR0 = K0/K2, VGPR1 = K1/K3
    const float* r1 = wb + 16 + 4 * col;          // k = col (0..15)
    v2f A1;
    A1.x = half ? r1[2] : r1[0];
    A1.y = half ? r1[3] : r1[1];

    int k2 = 16 + col;                            // k = 16..31 (>=27 padded)
    bool valid2 = (k2 < 27);
    const float* r2 = wb + 16 + 4 * (valid2 ? k2 : 0);
    v2f A2;
    A2.x = valid2 ? (half ? r2[2] : r2[0]) : 0.0f;
    A2.y = valid2 ? (half ? r2[3] : r2[1]) : (half ? 1e30f : 0.0f);

    // C fragment: C[M][N] = q_N  -> same value in all 8 VGPRs of a lane
    v8f C = { q, q, q, q, q, q, q, q };

    v8f D1 = __builtin_amdgcn_wmma_f32_16x16x4_f32(
        false, A1, false, Bf, (short)0, C, false, false);
    v8f D2 = __builtin_amdgcn_wmma_f32_16x16x4_f32(
        false, A2, false, Bf, (short)0, C, false, false);

    // In-lane min/argmin over the 16 rows this lane holds (first-occurrence ties)
    float best = 3.4e38f;
    int   bk   = 1 << 30;
#pragma unroll
    for (int r = 0; r < 8; ++r) {
        float v = D1[r];
        int   k = r + 8 * half;
        bool take = (v < best) || ((v == best) && (k < bk));
        best = take ? v : best;
        bk   = take ? k : bk;
    }
#pragma unroll
    for (int r = 0; r < 8; ++r) {
        int k = 16 + r + 8 * half;
        float v = D2[r];
        bool take = (k < 27) && ((v < best) || ((v == best) && (k < bk)));
        best = take ? v : best;
        bk   = take ? k : bk;
    }

    // Merge across wave halves (same column, complementary row sets)
    float ov = __shfl_xor(best, 16, 32);
    int   ok = __shfl_xor(bk, 16, 32);
    bool take = (ov < best) || ((ov == best) && (ok < bk));
    best = take ? ov : best;
    bk   = take ? ok : bk;

    if (half == 0) {
        float ox = (float)(bk / 9) - 1.0f;
        float oy = (float)((bk / 3) % 3) - 1.0f;
        float oz = (float)(bk % 3) - 1.0f;
        float mdx = dx + ox, mdy = dy + oy, mdz = dz + oz;
        float md2 = fmaxf(best, 0.0f);
        float md  = (md2 > 0.0f) ? sqrtf(md2) : 0.0f;
        float mdn = md / cell_scale;
        size_t o = (((size_t)b * NATOMS + n) * NATOMS + (size_t)(m0 + col)) * 6;
        out[o + 0] = mdx;
        out[o + 1] = mdy;
        out[o + 2] = mdz;
        out[o + 3] = md;
        out[o + 4] = md2;
        out[o + 5] = mdn;
    }
}

extern "C" void kernel_launch(void* const* d_in, const int* in_sizes, int n_in,
                              void* d_out, int out_size, void* d_ws, size_t ws_size,
                              hipStream_t stream)
{
    const float* coords  = (const float*)d_in[0];   // (B, 320, 3) f32
    const float* lattice = (const float*)d_in[1];   // (B, 6) f32
    float* out = (float*)d_out;                     // (B, 320, 320, 6) f32
    float* ws  = (float*)d_ws;

    int B = in_sizes[0] / (NATOMS * 3);             // = 8 for reference inputs

    geom_setup_kernel<<<B, 32, 0, stream>>>(lattice, ws, B);

    int totalWaves    = B * NATOMS * MTILES;        // one wave per 16-pair tile
    int wavesPerBlock = 8;                          // 256 threads = 8 wave32
    int blocks        = totalWaves / wavesPerBlock; // 6400 divides evenly
    geom_main_kernel<<<blocks, wavesPerBlock * 32, 0, stream>>>(coords, ws, out, B);
}